// DFMambaEncoder_20134806684290
// MI455X (gfx1250) — compile-verified
//
#include <hip/hip_runtime.h>
#include <hip/hip_bf16.h>

// ---------------------------------------------------------------------------
// DFMamba encoder for gfx1250 (MI455X).
//  - Dense 192-deep projections: V_WMMA_F32_16X16X4_F32, weights staged in LDS
//    (transposed, 196-float padded stride -> conflict-free ds_load_b64 A-frags).
//  - Selective scan: 16 states/channel in VGPRs; chunk staging of x_dbl and
//    the u tile through the Tensor Data Mover (tensor_load_to_lds +
//    s_wait_tensorcnt); direction outputs scatter-added with
//    global_atomic_add_f32 into the merged buffer.
// ---------------------------------------------------------------------------

typedef __attribute__((ext_vector_type(2))) float v2f;
typedef __attribute__((ext_vector_type(8))) float v8f;
typedef __attribute__((ext_vector_type(4))) unsigned int v4u;
typedef __attribute__((ext_vector_type(8))) int v8i;
typedef __attribute__((ext_vector_type(4))) int v4i;

#define D_INNER 192
#define D_STATE 16
#define DT_RANK 12
#define NROWS   44          // DT_RANK + 2*D_STATE
#define LLEN    4096
#define CHUNK   64
#define WPAD    196         // LDS row stride for weight tiles (bank spread)

static __device__ __forceinline__ v8f wmma4(v2f a, v2f b, v8f c) {
  // D = A(16x4 f32) * B(4x16 f32) + C(16x16 f32)
  return __builtin_amdgcn_wmma_f32_16x16x4_f32(
      /*neg_a=*/false, a, /*neg_b=*/false, b,
      /*c_mod=*/(short)0, c, /*reuse_a=*/false, /*reuse_b=*/false);
}

// direction permutation (transpose/flip compositions on the 64x64 grid)
static __device__ __forceinline__ int map_dir(int k, int t) {
  switch (k & 3) {
    case 0: return t;
    case 1: return ((t & 63) << 6) | (t >> 6);
    case 2: return 4095 - t;
    default: { int i = 4095 - t; return ((i & 63) << 6) | (i >> 6); }
  }
}

// ---------------------------------------------------------------------------
// Tensor Data Mover: 2D tile load Global -> LDS.  D# packed per CDNA5 ISA
// §8.3/§8.4.  dims/strides in elements (data_size = 4B), optional LDS row
// padding (pad codes per spec: interval 5 == 64 DWORDs, amount 0 == 1 DWORD).
// This toolchain uses the 6-arg builtin: (g0, g1, g2, g3, g4, cpol).
// ---------------------------------------------------------------------------
static __device__ __forceinline__ void tdm_load_2d(
    unsigned int lds_off, const void* gptr,
    unsigned int dim0, unsigned int dim1,
    unsigned int tile0, unsigned int tile1,
    unsigned long long stride0,
    int pad_enable, int pad_interval, int pad_amount) {
  const unsigned long long ga = (unsigned long long)gptr;
  v4u g0;
  g0.x = 1u;                                      // count=1, user descriptor
  g0.y = lds_off;                                 // lds_addr (bytes)
  g0.z = (unsigned int)ga;                        // global_addr[31:0]
  g0.w = (unsigned int)(ga >> 32) | (2u << 30);   // global_addr[56:32] | type=2
  v8i g1;
  unsigned int w0 = (2u << 16)                                    // data_size=4B
                  | (pad_enable ? (1u << 20) : 0u)
                  | (((unsigned int)pad_interval & 7u) << 22)
                  | (((unsigned int)pad_amount & 127u) << 25);
  g1[0] = (int)w0;                                               // wg_mask=0
  g1[1] = (int)((dim0 & 0xFFFFu) << 16);                          // abar=0|dim0.lo
  g1[2] = (int)(((dim0 >> 16) & 0xFFFFu) | ((dim1 & 0xFFFFu) << 16));
  g1[3] = (int)(((dim1 >> 16) & 0xFFFFu) | ((tile0 & 0xFFFFu) << 16));
  g1[4] = (int)(tile1 & 0xFFFFu);                                 // tile_dim2=0
  g1[5] = (int)(unsigned int)(stride0 & 0xFFFFFFFFull);
  g1[6] = (int)(unsigned int)((stride0 >> 32) & 0xFFFFull);       // stride1=0
  g1[7] = 0;
  const v4i gz4 = {0, 0, 0, 0};                   // 2D: groups 2/3 unused
  const v8i gz8 = {0, 0, 0, 0, 0, 0, 0, 0};
  __builtin_amdgcn_tensor_load_to_lds(g0, g1, gz4, gz4, gz8, 0);
}

// ---------------------------------------------------------------------------
// O[m,l] = sum_k W[k*192+m] * X[k*Ld+l], M=K=192.  One M-tile per block,
// 8 L-tiles (one per wave); W tile staged transposed in LDS.
// ---------------------------------------------------------------------------
__global__ __launch_bounds__(256)
void gemm192_wmma(const float* __restrict__ X, const float* __restrict__ W,
                  float* __restrict__ O, int Ld) {
  const int tM = blockIdx.x % 12;
  const int lb = blockIdx.x / 12;                 // gridDim.x = 12 * (Ld/128)
  const int m0 = tM * 16;
  const int lane = threadIdx.x & 31;
  const int wave = threadIdx.x >> 5;
  const int l0 = lb * 128 + wave * 16;
  const int half = lane >> 4, l15 = lane & 15;
  X += (size_t)blockIdx.z * D_INNER * (size_t)Ld;
  O += (size_t)blockIdx.z * D_INNER * (size_t)Ld;

  __shared__ float sW[16 * WPAD];                 // sW[m][k] transposed, padded
  for (int i = threadIdx.x; i < 16 * 192; i += 256) {
    const int mi = i & 15, k = i >> 4;            // coalesced over mi
    sW[mi * WPAD + k] = W[(size_t)k * 192 + m0 + mi];
  }
  __builtin_prefetch(&X[l0 + l15], 0, 1);         // global_prefetch_b8
  __syncthreads();

  v8f acc = {};
  for (int k = 0; k < 192; k += 4) {
    const int ka = k + 2 * half;
    v2f a, b;
    a.x = sW[l15 * WPAD + ka];                    // A[M=l15, K=ka]   (ds_load)
    a.y = sW[l15 * WPAD + ka + 1];
    b.x = X[(size_t)ka * Ld + l0 + l15];          // B[K=ka,   N=l15]
    b.y = X[(size_t)(ka + 1) * Ld + l0 + l15];
    acc = wmma4(a, b, acc);
  }
#pragma unroll
  for (int r = 0; r < 8; ++r) {
    const int m = m0 + r + 8 * half;              // C/D: vgpr r -> M=r / M=8+r
    O[(size_t)m * Ld + l0 + l15] = acc[r];
  }
}

// ---------------------------------------------------------------------------
// x_dbl[b,k][c,t] = sum_d Wp[k][c*192+d] * x_inner[b][d*L + map_dir(k,t)]
// c < 44 (padded to 48 -> 3 M-tiles).  Whole Wk staged in LDS per block.
// ---------------------------------------------------------------------------
__global__ __launch_bounds__(256)
void xdbl_wmma(const float* __restrict__ x_inner, const float* __restrict__ Wp,
               float* __restrict__ x_dbl) {
  const int z = blockIdx.z;                       // b*4 + k
  const int b = z >> 2, k = z & 3;
  const float* Xb = x_inner + (size_t)b * D_INNER * LLEN;
  const float* Wk = Wp + (size_t)k * NROWS * D_INNER;
  float* out = x_dbl + (size_t)z * NROWS * LLEN;

  __shared__ float sWk[NROWS * WPAD];             // sWk[c][d], padded stride
  for (int i = threadIdx.x; i < NROWS * 192; i += 256) {
    const int c = i / 192, dd = i - c * 192;      // coalesced over dd
    sWk[c * WPAD + dd] = Wk[(size_t)c * 192 + dd];
  }
  __syncthreads();

  const int lane = threadIdx.x & 31;
  const int wave = threadIdx.x >> 5;
  const int tile = blockIdx.x * 8 + wave;         // 3 * 256 tiles
  const int tM = tile % 3;
  const int tL = tile / 3;
  const int c0 = tM * 16, t0 = tL * 16;
  const int half = lane >> 4, l15 = lane & 15;
  const int src = map_dir(k, t0 + l15);           // gathered B column
  const int cA = c0 + l15;
  const bool av = (cA < NROWS);

  v8f acc = {};
  for (int d = 0; d < 192; d += 4) {
    const int da = d + 2 * half;
    v2f a, bf;
    a.x = av ? sWk[cA * WPAD + da] : 0.0f;
    a.y = av ? sWk[cA * WPAD + da + 1] : 0.0f;
    bf.x = Xb[(size_t)da * LLEN + src];
    bf.y = Xb[(size_t)(da + 1) * LLEN + src];
    acc = wmma4(a, bf, acc);
  }
#pragma unroll
  for (int r = 0; r < 8; ++r) {
    const int c = c0 + r + 8 * half;
    if (c < NROWS) out[(size_t)c * LLEN + t0 + l15] = acc[r];
  }
}

// ---------------------------------------------------------------------------
// Selective scan. One block per (b,k); thread = channel d; 16 states in VGPRs.
// Chunk staging via TDM for the contiguous directions; manual gather for the
// transposed ones.  dt-projection (rank 12) + softplus fused per step.
// ---------------------------------------------------------------------------
__global__ __launch_bounds__(192)
void scan_kernel(const float* __restrict__ x_dbl, const float* __restrict__ y_inner,
                 const float* __restrict__ dtw,   // [K,192,12]
                 const float* __restrict__ dtb,   // [K,192]
                 const float* __restrict__ A_logs,// [K*192,16]
                 const float* __restrict__ Ds,    // [K*192]
                 float* __restrict__ y_merged) {  // [B,192,L] (pre-zeroed)
  const int z = blockIdx.x;                       // b*4 + k
  const int b = z >> 2, k = z & 3;
  const int d = threadIdx.x;                      // 0..191
  const int kd = k * D_INNER + d;
  const float* xd = x_dbl + (size_t)z * NROWS * LLEN;
  const float* Yb = y_inner + (size_t)b * D_INNER * LLEN;
  float* ym = y_merged + (size_t)b * D_INNER * LLEN + (size_t)d * LLEN;

  __shared__ float s_xd[NROWS * CHUNK];
  __shared__ float s_u[D_INNER * (CHUNK + 1)];

  float h[D_STATE], Arow[D_STATE], w[DT_RANK];
#pragma unroll
  for (int n = 0; n < D_STATE; ++n) {
    h[n] = 0.0f;
    Arow[n] = -__expf(A_logs[(size_t)kd * D_STATE + n]);
  }
#pragma unroll
  for (int r = 0; r < DT_RANK; ++r) w[r] = dtw[(size_t)kd * DT_RANK + r];
  const float bias = dtb[kd];
  const float Dd = Ds[kd];

  for (int t0 = 0; t0 < LLEN; t0 += CHUNK) {
    __syncthreads();
    if (threadIdx.x == 0) {
      // x_dbl chunk: 44 rows x 64 cols, row stride 4096, dense in LDS
      tdm_load_2d((unsigned int)(size_t)&s_xd[0], xd + t0,
                  LLEN, NROWS, CHUNK, NROWS, LLEN, 0, 0, 0);
      if (!(k & 1)) {
        // u tile: 192 x 64 (k=0 forward; k=2 forward segment, read reversed);
        // pad 1 DWORD per 64 -> 65-float LDS rows.
        const int g0 = (k == 0) ? t0 : (LLEN - CHUNK - t0);
        tdm_load_2d((unsigned int)(size_t)&s_u[0], Yb + g0,
                    LLEN, D_INNER, CHUNK, D_INNER, LLEN, 1, 5, 0);
      }
    }
    if (k & 1) {                                  // transposed directions: gather
      for (int i = threadIdx.x; i < D_INNER * CHUNK; i += D_INNER) {
        const int dd = i / CHUNK, j = i - dd * CHUNK;
        s_u[dd * (CHUNK + 1) + j] = Yb[(size_t)dd * LLEN + map_dir(k, t0 + j)];
      }
    }
    if ((threadIdx.x >> 5) == 0) __builtin_amdgcn_s_wait_tensorcnt(0);
    __syncthreads();

    for (int j = 0; j < CHUNK; ++j) {
      float dt = bias;
#pragma unroll
      for (int r = 0; r < DT_RANK; ++r) dt += w[r] * s_xd[r * CHUNK + j];
      dt = (dt > 20.0f) ? dt : __logf(1.0f + __expf(dt));   // softplus
      const int jj = (k == 2) ? (CHUNK - 1 - j) : j;        // reversed segment
      const float u = s_u[d * (CHUNK + 1) + jj];
      const float du = dt * u;
      float y = Dd * u;
#pragma unroll
      for (int n = 0; n < D_STATE; ++n) {
        const float Bn = s_xd[(DT_RANK + n) * CHUNK + j];
        const float Cn = s_xd[(DT_RANK + D_STATE + n) * CHUNK + j];
        h[n] = __expf(dt * Arow[n]) * h[n] + du * Bn;
        y += h[n] * Cn;
      }
      atomicAdd(&ym[map_dir(k, t0 + j)], y);      // scatter into merged layout
    }
  }
}

// ---------------------------------------------------------------------------
__global__ void zero_kernel(float* __restrict__ p, int n) {
  const int i = blockIdx.x * blockDim.x + threadIdx.x;
  if (i < n) p[i] = 0.0f;
}

// LayerNorm over D=192 per (b,l); [B, D, L] layout (stride L).
__global__ __launch_bounds__(256)
void ln_kernel(const float* __restrict__ ym, const float* __restrict__ gamma,
               const float* __restrict__ beta, float* __restrict__ yn) {
  const int idx = blockIdx.x * blockDim.x + threadIdx.x;    // B*L threads
  const int b = idx >> 12, l = idx & (LLEN - 1);
  const float* p = ym + (size_t)b * D_INNER * LLEN + l;
  float s = 0.0f, s2 = 0.0f;
  for (int d = 0; d < D_INNER; ++d) {
    const float v = p[(size_t)d * LLEN];
    s += v; s2 += v * v;
  }
  const float mu = s * (1.0f / D_INNER);
  const float var = s2 * (1.0f / D_INNER) - mu * mu;
  const float inv = rsqrtf(var + 1e-5f);
  float* q = yn + (size_t)b * D_INNER * LLEN + l;
  for (int d = 0; d < D_INNER; ++d) {
    const float v = (p[(size_t)d * LLEN] - mu) * inv;
    q[(size_t)d * LLEN] = v * gamma[d] + beta[d];
  }
}

// ---------------------------------------------------------------------------
extern "C" void kernel_launch(void* const* d_in, const int* in_sizes, int n_in,
                              void* d_out, int out_size, void* d_ws, size_t ws_size,
                              hipStream_t stream) {
  const float* x      = (const float*)d_in[0];   // [4,192,64,64]
  const float* y      = (const float*)d_in[1];   // [4,192,64,64]
  const float* wpx    = (const float*)d_in[2];   // [192,192]
  const float* wpy    = (const float*)d_in[3];   // [192,192]
  const float* xproj  = (const float*)d_in[4];   // [4,44,192]
  const float* dtw    = (const float*)d_in[5];   // [4,192,12]
  const float* dtb    = (const float*)d_in[6];   // [4,192]
  const float* alogs  = (const float*)d_in[7];   // [768,16]
  const float* dsv    = (const float*)d_in[8];   // [768]
  const float* gamma  = (const float*)d_in[9];   // [192]
  const float* beta   = (const float*)d_in[10];  // [192]
  const float* wout   = (const float*)d_in[11];  // [192,192]

  const size_t PLANE = (size_t)4 * D_INNER * LLEN;         // 3,145,728 floats
  const size_t XDBL  = (size_t)16 * NROWS * LLEN;          // 2,883,584 floats
  float* ws        = (float*)d_ws;
  float* x_inner   = ws;                 // [B,192,L]
  float* y_inner   = x_inner + PLANE;    // [B,192,L]
  float* x_dbl     = y_inner + PLANE;    // [B*K,44,L]
  float* y_merged  = x_dbl + XDBL;       // [B,192,L]
  float* yn        = x_inner;            // reuse after scan

  dim3 blk(256);
  // 1) input projections (WMMA f32 GEMM; 12 M-tiles x 32 L-blocks; z=batch)
  gemm192_wmma<<<dim3(384, 1, 4), blk, 0, stream>>>(x, wpx, x_inner, LLEN);
  gemm192_wmma<<<dim3(384, 1, 4), blk, 0, stream>>>(y, wpy, y_inner, LLEN);
  // 2) x_dbl projection with direction gather
  xdbl_wmma<<<dim3(96, 1, 16), blk, 0, stream>>>(x_inner, xproj, x_dbl);
  // 3) zero the merge accumulator (harness poisons ws)
  zero_kernel<<<(int)((PLANE + 255) / 256), blk, 0, stream>>>(y_merged, (int)PLANE);
  // 4) selective scan over 4 directions (TDM staging, fused dt/softplus/merge)
  scan_kernel<<<16, 192, 0, stream>>>(x_dbl, y_inner, dtw, dtb, alogs, dsv, y_merged);
  // 5) layernorm
  ln_kernel<<<64, 256, 0, stream>>>(y_merged, gamma, beta, yn);
  // 6) output projection straight into d_out ([B,C,H,W] layout)
  gemm192_wmma<<<dim3(384, 1, 4), blk, 0, stream>>>(yn, wout, (float*)d_out, LLEN);
}